// LoopModel2_1889785610293
// MI455X (gfx1250) — compile-verified
//
#include <hip/hip_runtime.h>

typedef __attribute__((ext_vector_type(2))) float v2f;
typedef __attribute__((ext_vector_type(8))) float v8f;

// x: 8192 x 8192 fp32. out = x + y*(y-1)/2, computed as
// D = A(16x4, ones) * B(4x16, sum/4) + C(x tile) via V_WMMA_F32_16X16X4_F32.
// One wave32 per 16x16 tile; C/D loads/stores are coalesced dword accesses
// with immediate row offsets (row stride 8192 floats = 32768 bytes < 2^23).

#define MAT_W 8192

__global__ __launch_bounds__(256) void loop_add_wmma_kernel(
    const float* __restrict__ x,
    const int* __restrict__ yp,
    float* __restrict__ out,
    int tiles_x, int total_tiles) {

  const int lane   = threadIdx.x & 31;
  const int waveId = (blockIdx.x << 3) + (threadIdx.x >> 5);  // 8 waves / block
  if (waveId >= total_tiles) return;  // wave-uniform: EXEC stays all-ones

  // Scalar, wave-uniform read of the loop count; sum = y*(y-1)/2 (2016 for y=64).
  const int   y    = *yp;
  const float sum  = 0.5f * (float)y * (float)(y - 1);
  const float bv   = sum * 0.25f;  // exact /4; A*B contributes 4*bv = sum

  v2f a; a.x = 1.0f; a.y = 1.0f;   // A: 16x4 all ones (layout-agnostic)
  v2f b; b.x = bv;   b.y = bv;     // B: 4x16 all sum/4 (layout-agnostic)

  const int tm = waveId / tiles_x;
  const int tn = waveId - tm * tiles_x;

  // C/D layout: lane L, VGPR i  <->  row = 8*(L>=16) + i, col = L%16
  const int row0 = tm * 16 + ((lane >> 4) << 3);
  const int col  = tn * 16 + (lane & 15);

  const float* __restrict__ src = x   + (size_t)row0 * MAT_W + col;
  float*       __restrict__ dst = out + (size_t)row0 * MAT_W + col;

  v8f c;
#pragma unroll
  for (int i = 0; i < 8; ++i) {
    c[i] = src[(size_t)i * MAT_W];   // global_load_b32, ioffset = i*32768
  }

  // 8 args: (neg_a, A, neg_b, B, c_mod, C, reuse_a, reuse_b)
  v8f d = __builtin_amdgcn_wmma_f32_16x16x4_f32(
      /*neg_a=*/false, a, /*neg_b=*/false, b,
      /*c_mod=*/(short)0, c, /*reuse_a=*/false, /*reuse_b=*/false);

#pragma unroll
  for (int i = 0; i < 8; ++i) {
    dst[(size_t)i * MAT_W] = d[i];   // global_store_b32, ioffset = i*32768
  }
}

extern "C" void kernel_launch(void* const* d_in, const int* in_sizes, int n_in,
                              void* d_out, int out_size, void* d_ws, size_t ws_size,
                              hipStream_t stream) {
  const float* x  = (const float*)d_in[0];
  const int*   yp = (const int*)d_in[1];
  float*       out = (float*)d_out;

  const int n = in_sizes[0];          // 8192*8192
  const int H = n / MAT_W;            // 8192
  const int tiles_x = MAT_W / 16;     // 512
  const int tiles_y = H / 16;         // 512
  const int total_tiles = tiles_x * tiles_y;  // 262144 wave-tiles

  const int wavesPerBlock = 8;        // 256 threads = 8 wave32s
  const int blocks = (total_tiles + wavesPerBlock - 1) / wavesPerBlock;

  loop_add_wmma_kernel<<<blocks, 256, 0, stream>>>(x, yp, out, tiles_x, total_tiles);
}